// OnnxModule_57105885167965
// MI455X (gfx1250) — compile-verified
//
#include <hip/hip_runtime.h>
#include <math.h>

// MI455X (gfx1250): f32 GEMMs via V_WMMA_F32_16X16X4_F32 (wave32).
// Double-buffered LDS pipeline; global->LDS via async-LDS loads (ASYNCcnt)
// when the toolchain exposes them, else VGPR-staged copies.

typedef __attribute__((ext_vector_type(2))) float v2f;
typedef __attribute__((ext_vector_type(8))) float v8f;

#define BK 16
#define LDS_STRIDE 20  // BK+4: 16B-aligned rows, conflict-free b64 frag reads

#if defined(__has_builtin)
#if __has_builtin(__builtin_amdgcn_global_load_async_to_lds_b128) && \
    __has_builtin(__builtin_amdgcn_s_wait_asynccnt)
#define USE_ASYNC_LDS 1
#endif
#endif
#ifndef USE_ASYNC_LDS
#define USE_ASYNC_LDS 0
#endif

#if USE_ASYNC_LDS
// Builtin signature (from hipcc diagnostics): param 0 is a non-const
// vector_size(16) int pointer in the global (__device__/AS1) address space;
// param 1 is the LDS (AS3) pointer; then imm offset, imm cpol.
typedef int b128vec __attribute__((vector_size(16)));
#define AS_GLOBAL __attribute__((address_space(1)))
#define AS_LOCAL  __attribute__((address_space(3)))
#endif

__device__ __forceinline__ void wait_async_zero()
{
#if USE_ASYNC_LDS
    __builtin_amdgcn_s_wait_asynccnt(0);
#endif
}

// C[M x ldc] = act(A[M x lda] @ W[n_real x ldw]^T + bias)
// Grid: (Npad/TBN, M/TBM), 256 threads (8 waves). Cols >= n_real store 0.
template <int TBM, int TBN, int WAVES_M, int WAVES_N, bool GUARD_W>
__global__ __launch_bounds__(256)
void gemm_wmma_nt(const float* __restrict__ A, int lda,
                  const float* __restrict__ W, int ldw, int n_real,
                  const float* __restrict__ bias, int relu,
                  float* __restrict__ C, int ldc, int K)
{
    constexpr int RT = TBM / (16 * WAVES_M);   // 16-row tiles per wave
    constexpr int CT = TBN / (16 * WAVES_N);   // 16-col tiles per wave
    constexpr int A_PASSES = (TBM * BK) / 1024;
    constexpr int W_PASSES = (TBN * BK) / 1024;

    __shared__ float As[2][TBM * LDS_STRIDE];
    __shared__ float Ws[2][TBN * LDS_STRIDE];

    const int tid  = threadIdx.x;
    const int lane = tid & 31;
    const int wave = tid >> 5;             // 0..7
    const int wm   = wave % WAVES_M;
    const int wn   = wave / WAVES_M;
    const int half = lane >> 4;            // 0: lanes 0-15, 1: lanes 16-31
    const int ml   = lane & 15;

    const int row0 = blockIdx.y * TBM;
    const int col0 = blockIdx.x * TBN;

    const int r_ld = tid >> 2;             // tile row for pass 0 (+64 per pass)
    const int c_ld = (tid & 3) * 4;        // 0,4,8,12

    float4 astage[A_PASSES];
    float4 wstage[W_PASSES];

    // Issue the global fetch for chunk k0 into LDS buffer `buf`.
    auto issueA = [&](int buf, int k0) {
        #pragma unroll
        for (int i = 0; i < A_PASSES; ++i) {
            const int r = r_ld + i * 64;
            const float* src = A + (size_t)(row0 + r) * lda + k0 + c_ld;
#if USE_ASYNC_LDS
            __builtin_amdgcn_global_load_async_to_lds_b128(
                (AS_GLOBAL b128vec*)src,
                (AS_LOCAL b128vec*)&As[buf][r * LDS_STRIDE + c_ld], 0, 0);
#else
            astage[i] = *(const float4*)src;
#endif
        }
    };
    auto commitA = [&](int buf) {
#if !USE_ASYNC_LDS
        #pragma unroll
        for (int i = 0; i < A_PASSES; ++i) {
            const int r = r_ld + i * 64;
            *(float4*)&As[buf][r * LDS_STRIDE + c_ld] = astage[i];
        }
#endif
        (void)buf;
    };
    auto issueW = [&](int buf, int k0) {
        #pragma unroll
        for (int i = 0; i < W_PASSES; ++i) {
            const int r = r_ld + i * 64;
            const float* src = W + (size_t)(col0 + r) * ldw + k0 + c_ld;
            if constexpr (GUARD_W) {
                wstage[i] = make_float4(0.f, 0.f, 0.f, 0.f);
                if (col0 + r < n_real) wstage[i] = *(const float4*)src;
            } else {
#if USE_ASYNC_LDS
                __builtin_amdgcn_global_load_async_to_lds_b128(
                    (AS_GLOBAL b128vec*)src,
                    (AS_LOCAL b128vec*)&Ws[buf][r * LDS_STRIDE + c_ld], 0, 0);
#else
                wstage[i] = *(const float4*)src;
#endif
            }
        }
    };
    auto commitW = [&](int buf) {
        if constexpr (GUARD_W) {
            #pragma unroll
            for (int i = 0; i < W_PASSES; ++i) {
                const int r = r_ld + i * 64;
                *(float4*)&Ws[buf][r * LDS_STRIDE + c_ld] = wstage[i];
            }
        } else {
#if !USE_ASYNC_LDS
            #pragma unroll
            for (int i = 0; i < W_PASSES; ++i) {
                const int r = r_ld + i * 64;
                *(float4*)&Ws[buf][r * LDS_STRIDE + c_ld] = wstage[i];
            }
#endif
        }
        (void)buf;
    };

    v8f acc[RT][CT];
    #pragma unroll
    for (int rt = 0; rt < RT; ++rt)
        #pragma unroll
        for (int ct = 0; ct < CT; ++ct)
            acc[rt][ct] = (v8f){0.f,0.f,0.f,0.f,0.f,0.f,0.f,0.f};

    const int nchunks = K / BK;

    // Prologue: fill buffer 0.
    issueA(0, 0); issueW(0, 0);
    commitA(0);   commitW(0);
    wait_async_zero();
    __syncthreads();

    for (int ch = 0; ch < nchunks; ++ch) {
        const int cur = ch & 1, nxt = cur ^ 1;
        const bool more = (ch + 1 < nchunks);
        if (more) { issueA(nxt, (ch + 1) * BK); issueW(nxt, (ch + 1) * BK); }

        #pragma unroll
        for (int ks = 0; ks < 4; ++ks) {
            const int kc = ks * 4 + 2 * half;
            v2f a[RT], b[CT];
            #pragma unroll
            for (int rt = 0; rt < RT; ++rt)
                a[rt] = *(const v2f*)&As[cur][((wm * RT + rt) * 16 + ml) * LDS_STRIDE + kc];
            #pragma unroll
            for (int ct = 0; ct < CT; ++ct)
                b[ct] = *(const v2f*)&Ws[cur][((wn * CT + ct) * 16 + ml) * LDS_STRIDE + kc];
            #pragma unroll
            for (int rt = 0; rt < RT; ++rt)
                #pragma unroll
                for (int ct = 0; ct < CT; ++ct)
                    acc[rt][ct] = __builtin_amdgcn_wmma_f32_16x16x4_f32(
                        false, a[rt], false, b[ct], (short)0, acc[rt][ct],
                        false, false);
        }

        if (more) {
            commitA(nxt); commitW(nxt);
            wait_async_zero();
            __syncthreads();
        }
    }

    // C/D layout: VGPR r -> local row r + 8*half, lane ml -> col ml
    float bv[CT];
    #pragma unroll
    for (int ct = 0; ct < CT; ++ct) {
        const int ccol = col0 + (wn * CT + ct) * 16 + ml;
        bv[ct] = bias ? bias[ccol] : 0.f;
    }
    #pragma unroll
    for (int rt = 0; rt < RT; ++rt) {
        const int crow0 = row0 + (wm * RT + rt) * 16 + 8 * half;
        #pragma unroll
        for (int ct = 0; ct < CT; ++ct) {
            const int ccol = col0 + (wn * CT + ct) * 16 + ml;
            #pragma unroll
            for (int r = 0; r < 8; ++r) {
                float v = acc[rt][ct][r] + bv[ct];
                if (relu) v = fmaxf(v, 0.f);
                C[(size_t)(crow0 + r) * ldc + ccol] = v;
            }
        }
    }
}

// Per-row 3-class head: logits = h @ W2^T + b2, log_softmax, argmax.
__global__ __launch_bounds__(256)
void bio_head(const float* __restrict__ h, const float* __restrict__ W2,
              const float* __restrict__ b2, float* __restrict__ out_logits,
              int* __restrict__ pred, int rows)
{
    const int wave = threadIdx.x >> 5;
    const int lane = threadIdx.x & 31;
    const int row  = blockIdx.x * 8 + wave;
    if (row >= rows) return;
    const float* hr = h + (size_t)row * 768;
    float s0 = 0.f, s1 = 0.f, s2 = 0.f;
    for (int c = lane; c < 768; c += 32) {
        float v = hr[c];
        s0 += v * W2[c];
        s1 += v * W2[768 + c];
        s2 += v * W2[1536 + c];
    }
    for (int off = 16; off; off >>= 1) {
        s0 += __shfl_xor(s0, off, 32);
        s1 += __shfl_xor(s1, off, 32);
        s2 += __shfl_xor(s2, off, 32);
    }
    if (lane == 0) {
        float l0 = s0 + b2[0], l1 = s1 + b2[1], l2 = s2 + b2[2];
        float m = fmaxf(l0, fmaxf(l1, l2));
        float lse = m + logf(expf(l0 - m) + expf(l1 - m) + expf(l2 - m));
        out_logits[(size_t)row * 3 + 0] = l0 - lse;
        out_logits[(size_t)row * 3 + 1] = l1 - lse;
        out_logits[(size_t)row * 3 + 2] = l2 - lse;
        int p = 0; float best = l0;
        if (l1 > best) { best = l1; p = 1; }
        if (l2 > best) { p = 2; }
        pred[row] = p;
    }
}

// Batch-row compaction: dest[b] = prefix index among rows with any nonzero label.
__global__ void pool_prep(const int* __restrict__ labels, int* __restrict__ dest,
                          int Bn, int Tn)
{
    __shared__ int valid[64];
    const int b = threadIdx.x;
    if (b < Bn) {
        int v = 0;
        const int* lb = labels + (size_t)b * Tn;
        for (int t = 0; t < Tn; ++t) v |= (lb[t] != 0);
        valid[b] = v;
    }
    __syncthreads();
    if (threadIdx.x == 0) {
        int cnt = 0;
        for (int i = 0; i < Bn; ++i) dest[i] = valid[i] ? cnt++ : Bn;
    }
}

// Segment-mean pooling: one block per batch row; label==1 starts a segment,
// label!=0 tokens are averaged; segments compacted to the front.
__global__ __launch_bounds__(256)
void pool_segments(const int* __restrict__ labels, const float* __restrict__ hidden,
                   const int* __restrict__ dest, float* __restrict__ pooled,
                   int Tn, int Hd)
{
    __shared__ int slab[512];
    const int b  = blockIdx.x;
    const int ob = dest[b];
    const int tid = threadIdx.x;
    if (ob >= (int)gridDim.x) return;       // all-zero row: output stays zeroed
    for (int t = tid; t < Tn; t += 256) slab[t] = labels[(size_t)b * Tn + t];
    __syncthreads();
    const float* hb = hidden + (size_t)b * Tn * Hd;
    float* pb = pooled + (size_t)ob * Tn * Hd;
    const int h0 = tid, h1 = tid + 256, h2 = tid + 512;
    float a0 = 0.f, a1 = 0.f, a2 = 0.f;
    int cnt = 0, pos = 0;                    // uniform across block
    for (int t = 0; t < Tn; ++t) {
        const int lab = slab[t];
        if (lab == 1) {
            if (cnt > 0) {
                float fc = (float)cnt;
                float* dst = pb + (size_t)pos * Hd;
                dst[h0] = a0 / fc; dst[h1] = a1 / fc; dst[h2] = a2 / fc;
                ++pos;
            }
            const float* src = hb + (size_t)t * Hd;
            a0 = src[h0]; a1 = src[h1]; a2 = src[h2];
            cnt = 1;
        } else if (lab != 0) {
            const float* src = hb + (size_t)t * Hd;
            a0 += src[h0]; a1 += src[h1]; a2 += src[h2];
            ++cnt;
        }
    }
    if (cnt > 0) {
        float fc = (float)cnt;
        float* dst = pb + (size_t)pos * Hd;
        dst[h0] = a0 / fc; dst[h1] = a1 / fc; dst[h2] = a2 / fc;
    }
}

// In-place log_softmax over rows of 2048 (256 threads x 8 elems, held in regs).
__global__ __launch_bounds__(256)
void logsoftmax_rows(float* __restrict__ x, int ncols)
{
    __shared__ float red[8];
    float* row = x + (size_t)blockIdx.x * ncols;
    const int tid = threadIdx.x;
    const int wave = tid >> 5, lane = tid & 31;
    float v[8];
    float m = -INFINITY;
    #pragma unroll
    for (int i = 0; i < 8; ++i) { v[i] = row[tid + i * 256]; m = fmaxf(m, v[i]); }
    for (int off = 16; off; off >>= 1) m = fmaxf(m, __shfl_xor(m, off, 32));
    if (lane == 0) red[wave] = m;
    __syncthreads();
    if (wave == 0) {
        float t = (lane < 8) ? red[lane] : -INFINITY;
        for (int off = 4; off; off >>= 1) t = fmaxf(t, __shfl_xor(t, off, 32));
        if (lane == 0) red[0] = t;
    }
    __syncthreads();
    m = red[0];
    float s = 0.f;
    #pragma unroll
    for (int i = 0; i < 8; ++i) s += expf(v[i] - m);
    for (int off = 16; off; off >>= 1) s += __shfl_xor(s, off, 32);
    __syncthreads();
    if (lane == 0) red[wave] = s;
    __syncthreads();
    if (wave == 0) {
        float t = (lane < 8) ? red[lane] : 0.f;
        for (int off = 4; off; off >>= 1) t += __shfl_xor(t, off, 32);
        if (lane == 0) red[0] = t;
    }
    __syncthreads();
    const float lse = m + logf(red[0]);
    #pragma unroll
    for (int i = 0; i < 8; ++i) row[tid + i * 256] = v[i] - lse;
}

extern "C" void kernel_launch(void* const* d_in, const int* in_sizes, int n_in,
                              void* d_out, int out_size, void* d_ws, size_t ws_size,
                              hipStream_t stream)
{
    (void)in_sizes; (void)n_in; (void)out_size; (void)ws_size;
    const int*   labels = (const int*)d_in[0];
    const float* hidden = (const float*)d_in[1];
    const float* ent    = (const float*)d_in[2];
    const float* W1     = (const float*)d_in[3];
    const float* b1     = (const float*)d_in[4];
    const float* W2     = (const float*)d_in[5];
    const float* b2     = (const float*)d_in[6];
    const float* Wm     = (const float*)d_in[7];
    const float* Wd     = (const float*)d_in[8];

    const int B = 64, T = 512, H = 768, N = 2048, P = 300, Ppad = 320;
    const int M = B * T;  // 32768

    float* out_logits = (float*)d_out;
    float* out_s1 = out_logits + (size_t)M * 3;
    float* out_s2 = out_s1 + (size_t)M * N;

    char* ws = (char*)d_ws;
    float* buf_h  = (float*)ws;                                  // 32768x768 (also pooled)
    size_t off = (size_t)M * H * sizeof(float);
    float* buf_pm = (float*)(ws + off); off += (size_t)M * Ppad * sizeof(float);
    float* buf_pd = (float*)(ws + off); off += (size_t)N * Ppad * sizeof(float);
    int*   buf_pred = (int*)(ws + off); off += (size_t)M * sizeof(int);
    int*   buf_dest = (int*)(ws + off);

    const dim3 blk(256);

    // 1) h = relu(hidden @ W1^T + b1)   [32768x768, K=768] -- deep-tiled config
    gemm_wmma_nt<128, 128, 2, 4, false><<<dim3(H / 128, M / 128), blk, 0, stream>>>(
        hidden, H, W1, H, H, b1, 1, buf_h, H, H);
    // 2) 3-class head: logits (d_out) + predicted labels (ws)
    bio_head<<<dim3(M / 8), blk, 0, stream>>>(buf_h, W2, b2, out_logits, buf_pred, M);
    // 3) pd = ent @ Wd^T, padded to 320 cols (pad cols are exact zeros)
    gemm_wmma_nt<64, 64, 4, 2, true><<<dim3(Ppad / 64, N / 64), blk, 0, stream>>>(
        ent, H, Wd, H, P, nullptr, 0, buf_pd, Ppad, H);

    for (int pass = 0; pass < 2; ++pass) {
        float* Sout = pass ? out_s2 : out_s1;
        const int* lab = pass ? buf_pred : labels;
        (void)hipMemsetAsync(buf_h, 0, (size_t)M * H * sizeof(float), stream);  // pooled := 0
        pool_prep<<<1, 64, 0, stream>>>(lab, buf_dest, B, T);
        pool_segments<<<dim3(B), blk, 0, stream>>>(lab, hidden, buf_dest, buf_h, T, H);
        // pm = pooled @ Wm^T  (padded to 320)
        gemm_wmma_nt<64, 64, 4, 2, true><<<dim3(Ppad / 64, M / 64), blk, 0, stream>>>(
            buf_h, H, Wm, H, P, nullptr, 0, buf_pm, Ppad, H);
        // S = pm @ pd^T  (K=320 incl. zero pad -> exact) -- deep-tiled config
        gemm_wmma_nt<128, 128, 2, 4, false><<<dim3(N / 128, M / 128), blk, 0, stream>>>(
            buf_pm, Ppad, buf_pd, Ppad, N, nullptr, 0, Sout, N, Ppad);
        logsoftmax_rows<<<dim3(M), blk, 0, stream>>>(Sout, N);
    }
}